// StandGCN1_25056839205779
// MI455X (gfx1250) — compile-verified
//
#include <hip/hip_runtime.h>

typedef __attribute__((ext_vector_type(2))) float v2f;
typedef __attribute__((ext_vector_type(8))) float v8f;

// ---------------- utility kernels ----------------

__global__ void k_zero(float* __restrict__ p, long long n) {
    long long i = (long long)blockIdx.x * blockDim.x + threadIdx.x;
    if (i < n) p[i] = 0.0f;
}

__global__ void k_fill1(float* __restrict__ p, int n) {
    int i = blockIdx.x * blockDim.x + threadIdx.x;
    if (i < n) p[i] = 1.0f;
}

// deg[dst] += 1 per edge (self-loop already folded in via init to 1.0)
__global__ void k_degree(const long long* __restrict__ dst,
                         float* __restrict__ deg, long long E) {
    long long e = (long long)blockIdx.x * blockDim.x + threadIdx.x;
    if (e < E) atomicAdd(&deg[dst[e]], 1.0f);
}

__global__ void k_rsqrt(float* __restrict__ dinv, int n) {
    int i = blockIdx.x * blockDim.x + threadIdx.x;
    if (i < n) {
        float d = dinv[i];
        dinv[i] = (d > 0.0f) ? (1.0f / sqrtf(d)) : 0.0f;
    }
}

// ---------------- WMMA f32 GEMM (specialized): h[N,64] = x[N,256] @ W[256,64] --------
// Block = 256 threads = 8 waves; each wave owns a 16x16 tile (2 row tiles x 4 col tiles).
// W is staged into LDS once per block via gfx1250 async global->LDS copies, with the
// row stride padded to F_OUT+8 floats so both 16-lane halves of a wave read disjoint
// LDS bank groups. All inner-loop addresses are immediate offsets.
template <int F_IN, int F_OUT>
__global__ __launch_bounds__(256)
void k_gemm_wmma_lds(const float* __restrict__ x,
                     const float* __restrict__ W,
                     float* __restrict__ h, int N) {
    constexpr int LSTR = F_OUT + 8;              // 72 floats: bank-conflict-free padding
    __shared__ float wlds[F_IN * LSTR];          // 256*72*4 = 72 KB (<< 320 KB/WGP)

    // ---- stage W into LDS with async copies: 16B per thread per pass ----
    {
        const int t     = threadIdx.x;
        const int row   = t >> 4;                // 0..15
        const int chunk = t & 15;                // 0..15, 16 bytes each (64 floats/row)
#pragma unroll
        for (int p = 0; p < F_IN / 16; ++p) {
            const int r = p * 16 + row;
            unsigned lds_off = (unsigned)(size_t)(&wlds[r * LSTR]) + chunk * 16;
            const float* gaddr = W + (size_t)r * F_OUT + chunk * 4;
            asm volatile("global_load_async_to_lds_b128 %0, %1, off"
                         :: "v"(lds_off), "v"(gaddr) : "memory");
        }
        asm volatile("s_wait_asynccnt 0x0" ::: "memory");
    }
    __syncthreads();

    const int lane = threadIdx.x & 31;
    const int wave = threadIdx.x >> 5;
    const int lo   = lane & 15;
    const int hi   = lane >> 4;

    const int m0 = blockIdx.x * 32 + (wave >> 2) * 16;  // 16-row tile origin
    const int n0 = (wave & 3) * 16;                     // 16-col tile origin

    int row  = m0 + lo;
    int rowc = (row < N) ? row : (N - 1);               // clamp loads; stores guarded

    const float* __restrict__ xr   = x + (size_t)rowc * F_IN + 2 * hi;
    const float* __restrict__ wptr = &wlds[2 * hi * LSTR + n0 + lo];

    v8f c = {};
#pragma unroll
    for (int k0 = 0; k0 < F_IN; k0 += 4) {
        v2f a;
        a.x = xr[k0];                       // A[M=lo][K = k0 + 2*hi + 0]
        a.y = xr[k0 + 1];                   // A[M=lo][K = k0 + 2*hi + 1]
        v2f bfrag;
        bfrag.x = wptr[k0 * LSTR];          // B[K = k0+2*hi  ][N = n0+lo]
        bfrag.y = wptr[(k0 + 1) * LSTR];    // B[K = k0+2*hi+1][N = n0+lo]
        c = __builtin_amdgcn_wmma_f32_16x16x4_f32(
                false, a, false, bfrag, (short)0, c, false, false);
    }

#pragma unroll
    for (int v = 0; v < 8; ++v) {
        int r = m0 + v + 8 * hi;
        if (r < N) h[(size_t)r * F_OUT + n0 + lo] = c[v];
    }
}

// ---------------- generic fallback (runtime shapes) ----------------
__global__ void k_gemm_wmma_generic(const float* __restrict__ x,
                                    const float* __restrict__ W,
                                    float* __restrict__ h,
                                    int N, int F_IN, int F_OUT) {
    const int lane = threadIdx.x & 31;
    const int wave = threadIdx.x >> 5;
    const int lo   = lane & 15;
    const int hi   = lane >> 4;

    const int m0 = blockIdx.x * 32 + (wave >> 2) * 16;
    const int n0 = (wave & 3) * 16;
    if (n0 >= F_OUT) return;

    int row  = m0 + lo;
    int rowc = (row < N) ? row : (N - 1);
    const float* __restrict__ xr = x + (size_t)rowc * F_IN + 2 * hi;

    v8f c = {};
    for (int k0 = 0; k0 < F_IN; k0 += 4) {
        v2f a;
        a.x = xr[k0];
        a.y = xr[k0 + 1];
        const int kb = k0 + 2 * hi;
        v2f bfrag;
        bfrag.x = W[(size_t)kb * F_OUT + n0 + lo];
        bfrag.y = W[(size_t)(kb + 1) * F_OUT + n0 + lo];
        c = __builtin_amdgcn_wmma_f32_16x16x4_f32(
                false, a, false, bfrag, (short)0, c, false, false);
    }
#pragma unroll
    for (int v = 0; v < 8; ++v) {
        int r = m0 + v + 8 * hi;
        if (r < N) h[(size_t)r * F_OUT + n0 + lo] = c[v];
    }
}

// ---------------- edge scatter: out[dst] += h[src] * dinv[src]*dinv[dst] ----------------
// One wave per edge (index forced into SGPRs -> scalar loads of src/dst/dinv);
// each lane handles 2 features: coalesced float2 gather of h + two f32 atomics.
__global__ __launch_bounds__(256)
void k_scatter(const long long* __restrict__ src,
               const long long* __restrict__ dst,
               const float* __restrict__ h,
               const float* __restrict__ dinv,
               float* __restrict__ out,
               long long E) {
    int wv = __builtin_amdgcn_readfirstlane((int)(threadIdx.x >> 5));  // wave-uniform
    long long e = (long long)blockIdx.x * 8 + wv;
    const int lane = threadIdx.x & 31;
    if (e >= E) return;
    long long s = src[e];   // scalar loads (uniform index)
    long long d = dst[e];
    float w = dinv[s] * dinv[d];
    const float2* __restrict__ hp = (const float2*)(h + s * 64);
    float2 v = hp[lane];
    float* op = out + d * 64 + 2 * lane;
    atomicAdd(op,     v.x * w);
    atomicAdd(op + 1, v.y * w);
}

// ---------------- finalize: self-loop term + bias ----------------
__global__ void k_finalize(float* __restrict__ out,
                           const float* __restrict__ h,
                           const float* __restrict__ dinv,
                           const float* __restrict__ b,
                           long long total) {
    long long gid = (long long)blockIdx.x * blockDim.x + threadIdx.x;
    if (gid < total) {
        long long i = gid >> 6;   // F_OUT == 64
        int f = (int)(gid & 63);
        float di = dinv[i];
        out[gid] += h[gid] * di * di + b[f];
    }
}

// ---------------- launcher ----------------
extern "C" void kernel_launch(void* const* d_in, const int* in_sizes, int n_in,
                              void* d_out, int out_size, void* d_ws, size_t ws_size,
                              hipStream_t stream) {
    const float*     x   = (const float*)d_in[0];
    const long long* adj = (const long long*)d_in[1];  // [2, E] int64, row-major
    const float*     W   = (const float*)d_in[2];
    const float*     b   = (const float*)d_in[3];
    float*           out = (float*)d_out;

    const int F_OUT = in_sizes[3];                 // 64
    const int F_IN  = in_sizes[2] / F_OUT;         // 256
    const int N     = in_sizes[0] / F_IN;          // 100000
    const long long E = (long long)in_sizes[1] / 2;

    const long long* src = adj;
    const long long* dst = adj + E;

    // workspace layout: h [N*F_OUT floats] then dinv [N floats]
    float* h    = (float*)d_ws;
    float* dinv = h + (size_t)N * F_OUT;

    const long long totOut = (long long)N * F_OUT;

    k_zero<<<(int)((totOut + 255) / 256), 256, 0, stream>>>(out, totOut);
    k_fill1<<<(N + 255) / 256, 256, 0, stream>>>(dinv, N);            // deg=1 (self-loop)
    k_degree<<<(int)((E + 255) / 256), 256, 0, stream>>>(dst, dinv, E);
    k_rsqrt<<<(N + 255) / 256, 256, 0, stream>>>(dinv, N);

    if (F_IN == 256 && F_OUT == 64) {
        k_gemm_wmma_lds<256, 64><<<(N + 31) / 32, 256, 0, stream>>>(x, W, h, N);
    } else {
        k_gemm_wmma_generic<<<(N + 31) / 32, 256, 0, stream>>>(x, W, h, N, F_IN, F_OUT);
    }

    const long long totScatter = E * 32;
    k_scatter<<<(int)((totScatter + 255) / 256), 256, 0, stream>>>(src, dst, h, dinv, out, E);

    k_finalize<<<(int)((totOut + 255) / 256), 256, 0, stream>>>(out, h, dinv, b, totOut);
}